// GNNEncoder_1906965479432
// MI455X (gfx1250) — compile-verified
//
#include <hip/hip_runtime.h>

// GraphSAGE 2-layer encoder for MI455X (gfx1250, wave32).
//   layer(x): relu?( segmean(x[src]->dst) @ Wl^T + b + x @ Wr^T )
//
// Roofline: GEMMs total ~21 GFLOP (microseconds on the matrix pipe); the
// edge aggregation (~655 MB of L2-resident gather + f32 atomic scatter per
// layer) dominates. So GEMMs stay fp32 (V_WMMA_F32_16X16X4_F32) for full
// reference precision — bf16 would buy nothing end-to-end.
//
// GEMM shape: each wave32 owns 5 row tiles x 1 col tile (80x16 of C).
// Weight fragments are loaded once per k-step and reused across the 5 row
// tiles: 12 b64 loads per 10 WMMAs (~6.7 FLOP/B from cache). The 1/deg mean
// scale is applied to the aggregation accumulator in the epilogue (it
// commutes with the GEMM row-wise), keeping the inner loop pure WMMA.

#define NN 10000      // nodes
#define DD 512        // feature dim (in = hid = out)
#define NE 160000     // edges
#define MT 5          // row tiles per wave (625 row tiles / 5 = 125 exact)

typedef __attribute__((ext_vector_type(2))) float v2f;
typedef __attribute__((ext_vector_type(8))) float v8f;

__global__ void zero_kernel(float* __restrict__ p, long long n) {
  long long i = (long long)blockIdx.x * blockDim.x + threadIdx.x;
  long long stride = (long long)gridDim.x * blockDim.x;
  for (; i < n; i += stride) p[i] = 0.0f;
}

__global__ void count_kernel(const int* __restrict__ dst, float* __restrict__ cnt) {
  int i = blockIdx.x * blockDim.x + threadIdx.x;
  if (i < NE) atomicAdd(&cnt[dst[i]], 1.0f);
}

__global__ void recip_kernel(float* __restrict__ cnt) {
  int i = blockIdx.x * blockDim.x + threadIdx.x;
  if (i < NN) cnt[i] = 1.0f / fmaxf(cnt[i], 1.0f);
}

// One block (128 threads) per edge: coalesced 2KB row gather + non-returning
// GLOBAL_ATOMIC_ADD_F32 scatter into the L2-resident accumulator.
__global__ __launch_bounds__(128) void scatter_kernel(
    const float* __restrict__ feat, const int* __restrict__ src,
    const int* __restrict__ dst, float* __restrict__ agg) {
  const int e = blockIdx.x;
  const int s = src[e];   // uniform -> scalar load
  const int d = dst[e];
  const float4 v = ((const float4*)(feat + (size_t)s * DD))[threadIdx.x];
  float* ap = agg + (size_t)d * DD + (size_t)threadIdx.x * 4;
  atomicAdd(ap + 0, v.x);
  atomicAdd(ap + 1, v.y);
  atomicAdd(ap + 2, v.z);
  atomicAdd(ap + 3, v.w);
}

// out = (Agg @ Wl^T) * inv[row] + bias + Xin @ Wr^T,  optional ReLU.
// fp32 16x16x4 fragment layout (ISA 7.12.2), lane = 16*half + r:
//   A (16x4): lane holds row r; VGPR v holds K = 2*half + v
//   B (4x16): lane holds col r; VGPR v holds K = 2*half + v   (B row = W row `col`)
//   C (16x16): VGPR v holds row v + 8*half, col = r
template <bool RELU>
__global__ __launch_bounds__(256) void sage_gemm(
    const float* __restrict__ Agg, const float* __restrict__ inv,
    const float* __restrict__ Xin, const float* __restrict__ Wl,
    const float* __restrict__ Wr, const float* __restrict__ bias,
    float* __restrict__ out) {
  const int wave = threadIdx.x >> 5;
  const int lane = threadIdx.x & 31;
  const int half = lane >> 4;
  const int r    = lane & 15;
  const int idx  = blockIdx.x * 8 + wave;      // 500 blocks * 8 waves = 4000 jobs
  const int rg   = idx >> 5;                   // 125 groups of 80 rows
  const int ct   = idx & 31;                   // 32 col tiles (512/16 exact)
  const int row0 = rg * (16 * MT) + r;
  const int col  = ct * 16 + r;

  // Base pointers; per-tile row offsets become immediate instruction offsets.
  const float* a1p = Agg + (size_t)row0 * DD + 2 * half;
  const float* a2p = Xin + (size_t)row0 * DD + 2 * half;
  const float* blp = Wl  + (size_t)col  * DD + 2 * half;
  const float* brp = Wr  + (size_t)col  * DD + 2 * half;

  v8f c1[MT] = {};   // Agg @ Wl^T accumulators (scaled by inv in epilogue)
  v8f c2[MT] = {};   // Xin @ Wr^T accumulators
#pragma unroll 2
  for (int k = 0; k < DD; k += 4) {
    const v2f fbl = *(const v2f*)(blp + k);    // weight frags shared by MT tiles
    const v2f fbr = *(const v2f*)(brp + k);
    __builtin_prefetch(blp + k + 64, 0, 3);    // near-level global_prefetch_b8
    __builtin_prefetch(brp + k + 64, 0, 3);
#pragma unroll
    for (int t = 0; t < MT; ++t) {
      const v2f fa1 = *(const v2f*)(a1p + k + t * (16 * DD));
      const v2f fa2 = *(const v2f*)(a2p + k + t * (16 * DD));
      c1[t] = __builtin_amdgcn_wmma_f32_16x16x4_f32(false, fa1, false, fbl,
                                                    (short)0, c1[t], false, false);
      c2[t] = __builtin_amdgcn_wmma_f32_16x16x4_f32(false, fa2, false, fbr,
                                                    (short)0, c2[t], false, false);
    }
  }

  const float b = bias[col];                   // bias per output column = lane's col
#pragma unroll
  for (int t = 0; t < MT; ++t) {
    const int rbase = rg * (16 * MT) + 16 * t + 8 * half;
    float* op = out + (size_t)rbase * DD + (size_t)ct * 16 + r;
#pragma unroll
    for (int v = 0; v < 8; ++v) {
      float val = c1[t][v] * inv[rbase + v] + c2[t][v] + b;  // mean-scale here
      if (RELU) val = fmaxf(val, 0.0f);
      op[(size_t)v * DD] = val;                // C row = rbase + v
    }
  }
}

extern "C" void kernel_launch(void* const* d_in, const int* in_sizes, int n_in,
                              void* d_out, int out_size, void* d_ws, size_t ws_size,
                              hipStream_t stream) {
  const float* x   = (const float*)d_in[0];
  const int*   ei  = (const int*)d_in[1];     // edge_index [2, NE]: row0=src, row1=dst
  const int*   src = ei;
  const int*   dst = ei + NE;
  const float* W1l = (const float*)d_in[2];
  const float* b1  = (const float*)d_in[3];
  const float* W1r = (const float*)d_in[4];
  const float* W2l = (const float*)d_in[5];
  const float* b2  = (const float*)d_in[6];
  const float* W2r = (const float*)d_in[7];
  float* out = (float*)d_out;

  // Workspace: agg[N*D] | h[N*D] | cnt[N]  (~41 MB, fits MI455X 192 MB L2)
  float* agg = (float*)d_ws;
  float* h   = agg + (size_t)NN * DD;
  float* cnt = h   + (size_t)NN * DD;

  // ---- layer 1 ----
  zero_kernel<<<2048, 256, 0, stream>>>(agg, (long long)NN * DD);
  zero_kernel<<<64,   256, 0, stream>>>(cnt, (long long)NN);
  count_kernel<<<(NE + 255) / 256, 256, 0, stream>>>(dst, cnt);
  recip_kernel<<<(NN + 255) / 256, 256, 0, stream>>>(cnt);   // cnt <- 1/max(deg,1)
  scatter_kernel<<<NE, 128, 0, stream>>>(x, src, dst, agg);
  sage_gemm<true><<<500, 256, 0, stream>>>(agg, cnt, x, W1l, W1r, b1, h);

  // ---- layer 2 (degrees unchanged; reuse cnt) ----
  zero_kernel<<<2048, 256, 0, stream>>>(agg, (long long)NN * DD);
  scatter_kernel<<<NE, 128, 0, stream>>>(h, src, dst, agg);
  sage_gemm<false><<<500, 256, 0, stream>>>(agg, cnt, h, W2l, W2r, b2, out);
}